// DistillationLoss_1211180778390
// MI455X (gfx1250) — compile-verified
//
#include <hip/hip_runtime.h>
#include <math.h>

typedef __attribute__((ext_vector_type(2))) float v2f;
typedef __attribute__((ext_vector_type(8))) float v8f;

#define ALPHA_W 0.7f
#define EPS_N 1e-8f

// ---------------- block reduction helper (blockDim must be 256) ----------------
__device__ __forceinline__ float blockReduceSum256(float v) {
    __shared__ float sbuf[256];
    int t = threadIdx.x;
    sbuf[t] = v;
    __syncthreads();
    for (int off = 128; off > 0; off >>= 1) {
        if (t < off) sbuf[t] += sbuf[t + off];
        __syncthreads();
    }
    return sbuf[0];
}

// ---------------- zero init ----------------
__global__ void zero_kernel(float* p, int n) {
    int i = blockIdx.x * blockDim.x + threadIdx.x;
    if (i < n) p[i] = 0.0f;
}

// ---------------- WMMA brute-force row-min of cdist^2 ----------------
// One wave owns a 16-row strip of A and scans all of B in 16-column tiles,
// 4 tiles per iteration (software pipelined: loads batched ahead of the
// 4 back-to-back v_wmma_f32_16x16x4_f32 ops). Non-argmin path uses fminf
// so the scan update lowers to a single v_min_f32 per row.
template <bool ARGMIN>
__global__ void cdist_rowmin_kernel(const float* __restrict__ A, int nA,
                                    const float* __restrict__ B, int nB,
                                    float* __restrict__ rowMin,
                                    int* __restrict__ rowArg) {
    const int lane  = threadIdx.x & 31;
    const int wave  = threadIdx.x >> 5;
    const int strip = blockIdx.x * (blockDim.x >> 5) + wave;
    const int row0  = strip * 16;
    if (row0 >= nA) return;            // wave-uniform

    const int nl   = lane & 15;
    const int half = lane >> 4;

    // A operand: lanes 0-15 supply (x,y) as K=0,1; lanes 16-31 supply (z,0) as K=2,3
    int ra = row0 + nl;
    if (ra >= nA) ra = nA - 1;         // duplicate clamp: min unaffected
    const float ax = A[3 * ra + 0], ay = A[3 * ra + 1], az = A[3 * ra + 2];
    const float na_own = ax * ax + ay * ay + az * az;
    v2f a;
    a.x = half ? az : ax;
    a.y = half ? 0.0f : ay;

    // broadcast |a|^2 for the 8 rows this half-wave's D registers cover
    float na8[8];
#pragma unroll
    for (int r = 0; r < 8; ++r)
        na8[r] = __shfl(na_own, r + 8 * half, 32);

    float minv[8];
    int   minc[8];
#pragma unroll
    for (int r = 0; r < 8; ++r) { minv[r] = 3.0e38f; minc[r] = 0; }

    int j0 = 0;
    // main loop: 4 column tiles (64 columns) per iteration
    for (; j0 + 64 <= nB; j0 += 64) {
        float bx[4], by[4], bz[4];
#pragma unroll
        for (int u = 0; u < 4; ++u) {
            const int cb = j0 + 16 * u + nl;
            bx[u] = B[3 * cb + 0];
            by[u] = B[3 * cb + 1];
            bz[u] = B[3 * cb + 2];
        }
#pragma unroll
        for (int u = 0; u < 4; ++u) {
            const int   cb     = j0 + 16 * u + nl;
            const float nb_own = bx[u] * bx[u] + by[u] * by[u] + bz[u] * bz[u];
            v2f b;
            b.x = half ? bz[u] : bx[u];
            b.y = half ? 0.0f : by[u];

            v8f c = {0.f, 0.f, 0.f, 0.f, 0.f, 0.f, 0.f, 0.f};
            v8f d = __builtin_amdgcn_wmma_f32_16x16x4_f32(
                false, a, false, b, (short)0, c, false, false);

#pragma unroll
            for (int r = 0; r < 8; ++r) {
                const float d2 = na8[r] + nb_own - 2.0f * d[r];
                if (ARGMIN) {
                    if (d2 < minv[r]) { minv[r] = d2; minc[r] = cb; }
                } else {
                    minv[r] = fminf(minv[r], d2);   // single v_min_f32
                }
            }
        }
    }
    // tail: one 16-column tile at a time, clamped
    for (; j0 < nB; j0 += 16) {
        int cb = j0 + nl;
        if (cb >= nB) cb = nB - 1;
        const float bx = B[3 * cb + 0], by = B[3 * cb + 1], bz = B[3 * cb + 2];
        const float nb_own = bx * bx + by * by + bz * bz;
        v2f b;
        b.x = half ? bz : bx;
        b.y = half ? 0.0f : by;

        v8f c = {0.f, 0.f, 0.f, 0.f, 0.f, 0.f, 0.f, 0.f};
        v8f d = __builtin_amdgcn_wmma_f32_16x16x4_f32(
            false, a, false, b, (short)0, c, false, false);

#pragma unroll
        for (int r = 0; r < 8; ++r) {
            const float d2 = na8[r] + nb_own - 2.0f * d[r];
            if (ARGMIN) {
                if (d2 < minv[r]) { minv[r] = d2; minc[r] = cb; }
            } else {
                minv[r] = fminf(minv[r], d2);
            }
        }
    }

    // min-reduce across the 16 lanes of each half (xor masks stay inside halves)
#pragma unroll
    for (int m = 1; m < 16; m <<= 1) {
#pragma unroll
        for (int r = 0; r < 8; ++r) {
            const float ov = __shfl_xor(minv[r], m, 32);
            if (ARGMIN) {
                const int oc = __shfl_xor(minc[r], m, 32);
                if (ov < minv[r] || (ov == minv[r] && oc < minc[r])) {
                    minv[r] = ov;
                    minc[r] = oc;
                }
            } else {
                minv[r] = fminf(minv[r], ov);
            }
        }
    }

    if (nl == 0) {
        const int base = row0 + 8 * half;
#pragma unroll
        for (int r = 0; r < 8; ++r) {
            if (base + r < nA) {
                rowMin[base + r] = minv[r];
                if (ARGMIN) rowArg[base + r] = minc[r];
            }
        }
    }
}

// ---------------- sum of sqrt(max(x,0)) into accum slot ----------------
__global__ void sum_sqrt_kernel(const float* __restrict__ x, int n, float* slot) {
    float v = 0.0f;
    for (int i = blockIdx.x * blockDim.x + threadIdx.x; i < n;
         i += gridDim.x * blockDim.x)
        v += sqrtf(fmaxf(x[i], 0.0f));
    v = blockReduceSum256(v);
    if (threadIdx.x == 0) atomicAdd(slot, v);
}

// ---------------- edge loss + adjacency accumulation ----------------
__device__ __forceinline__ void edge_accum(int ia, int ib,
                                           float ax, float ay, float az,
                                           float bx, float by, float bz,
                                           float* nbr, float* deg, float& local) {
    const float dx = ax - bx, dy = ay - by, dz = az - bz;
    local += dx * dx + dy * dy + dz * dz;
    atomicAdd(&nbr[3 * ia + 0], bx);
    atomicAdd(&nbr[3 * ia + 1], by);
    atomicAdd(&nbr[3 * ia + 2], bz);
    atomicAdd(&deg[ia], 1.0f);
    atomicAdd(&nbr[3 * ib + 0], ax);
    atomicAdd(&nbr[3 * ib + 1], ay);
    atomicAdd(&nbr[3 * ib + 2], az);
    atomicAdd(&deg[ib], 1.0f);
}

__global__ void edge_kernel(const float* __restrict__ V,
                            const int* __restrict__ faces, int F,
                            float* nbr, float* deg, float* slot) {
    const int f = blockIdx.x * blockDim.x + threadIdx.x;
    float local = 0.0f;
    if (f < F) {
        const int i0 = faces[3 * f + 0];
        const int i1 = faces[3 * f + 1];
        const int i2 = faces[3 * f + 2];
        const float x0 = V[3 * i0], y0 = V[3 * i0 + 1], z0 = V[3 * i0 + 2];
        const float x1 = V[3 * i1], y1 = V[3 * i1 + 1], z1 = V[3 * i1 + 2];
        const float x2 = V[3 * i2], y2 = V[3 * i2 + 1], z2 = V[3 * i2 + 2];
        edge_accum(i0, i1, x0, y0, z0, x1, y1, z1, nbr, deg, local);
        edge_accum(i1, i2, x1, y1, z1, x2, y2, z2, nbr, deg, local);
        edge_accum(i2, i0, x2, y2, z2, x0, y0, z0, nbr, deg, local);
    }
    local = blockReduceSum256(local);
    if (threadIdx.x == 0) atomicAdd(slot, local);
}

// ---------------- laplacian smoothing ----------------
__global__ void smooth_kernel(const float* __restrict__ V,
                              const float* __restrict__ nbr,
                              const float* __restrict__ deg, int N, float* slot) {
    const int i = blockIdx.x * blockDim.x + threadIdx.x;
    float s = 0.0f;
    if (i < N) {
        const float dg = fmaxf(deg[i], 1.0f);
        const float lx = nbr[3 * i + 0] / dg - V[3 * i + 0];
        const float ly = nbr[3 * i + 1] / dg - V[3 * i + 1];
        const float lz = nbr[3 * i + 2] / dg - V[3 * i + 2];
        s = sqrtf(lx * lx + ly * ly + lz * lz);
    }
    s = blockReduceSum256(s);
    if (threadIdx.x == 0) atomicAdd(slot, s);
}

// ---------------- normal cosine via nearest teacher vertex ----------------
__global__ void normal_kernel(const float* __restrict__ sn,
                              const float* __restrict__ tn,
                              const int* __restrict__ nearest, int N, float* slot) {
    const int i = blockIdx.x * blockDim.x + threadIdx.x;
    float c = 0.0f;
    if (i < N) {
        const int j = nearest[i];
        const float sx = sn[3 * i], sy = sn[3 * i + 1], sz = sn[3 * i + 2];
        const float tx = tn[3 * j], ty = tn[3 * j + 1], tz = tn[3 * j + 2];
        const float num = sx * tx + sy * ty + sz * tz;
        const float den = fmaxf(sqrtf(sx * sx + sy * sy + sz * sz), EPS_N) *
                          fmaxf(sqrtf(tx * tx + ty * ty + tz * tz), EPS_N);
        c = num / den;
    }
    c = blockReduceSum256(c);
    if (threadIdx.x == 0) atomicAdd(slot, c);
}

// ---------------- final scalar composition ----------------
__global__ void finalize_kernel(const float* __restrict__ accum,
                                int N, int P, int G, int F, float* out) {
    const float loss_teacher = 0.5f * (accum[0] / (float)N + accum[1] / (float)P);
    const float loss_gt      = 0.5f * (accum[2] / (float)N + accum[3] / (float)G);
    const float loss_chamfer = ALPHA_W * loss_teacher + (1.0f - ALPHA_W) * loss_gt;
    const float loss_edge    = accum[4] / (3.0f * (float)F);
    const float loss_smooth  = accum[5] / (float)N;
    const float loss_normal  = 1.0f - accum[6] / (float)N;
    out[0] = 1.0f * loss_chamfer + 2.0f * loss_edge + 1.0f * loss_smooth +
             0.5f * loss_normal;
}

// ---------------- launcher ----------------
extern "C" void kernel_launch(void* const* d_in, const int* in_sizes, int n_in,
                              void* d_out, int out_size, void* d_ws, size_t ws_size,
                              hipStream_t stream) {
    const float* sv    = (const float*)d_in[0];  // student_verts   (N,3)
    const float* snrm  = (const float*)d_in[1];  // student_normals (N,3)
    const float* tv    = (const float*)d_in[2];  // teacher_verts   (M,3)
    const float* tnrm  = (const float*)d_in[3];  // teacher_normals (M,3)
    const float* tp    = (const float*)d_in[4];  // teacher_points  (P,3)
    const float* gp    = (const float*)d_in[5];  // gt_points       (G,3)
    const int*   faces = (const int*)d_in[6];    // faces           (F,3)

    const int N = in_sizes[0] / 3;
    const int M = in_sizes[2] / 3;
    const int P = in_sizes[4] / 3;
    const int G = in_sizes[5] / 3;
    const int F = in_sizes[6] / 3;

    float* ws     = (float*)d_ws;
    float* accum  = ws;                 // 8 floats
    float* nbr    = ws + 8;             // 3N
    float* deg    = nbr + 3 * N;        // N
    float* m_s_tp = deg + N;            // N
    float* m_tp_s = m_s_tp + N;         // P
    float* m_s_gt = m_tp_s + P;         // N
    float* m_gt_s = m_s_gt + N;         // G
    float* m_s_tv = m_gt_s + G;         // N (row mins vs teacher verts, value unused)
    int*   nearest = (int*)(m_s_tv + N);// N ints

    // zero: accum(8) + nbr(3N) + deg(N)  (contiguous)
    const int zn = 8 + 4 * N;
    zero_kernel<<<(zn + 255) / 256, 256, 0, stream>>>(ws, zn);

    auto strips = [](int n) { return (n + 15) / 16; };
    const int BD = 128;  // 4 waves/block, 1 strip per wave

    // chamfer halves (row mins both directions; swap avoids cross-block col-min)
    cdist_rowmin_kernel<false><<<(strips(N) + 3) / 4, BD, 0, stream>>>(sv, N, tp, P, m_s_tp, nullptr);
    cdist_rowmin_kernel<false><<<(strips(P) + 3) / 4, BD, 0, stream>>>(tp, P, sv, N, m_tp_s, nullptr);
    cdist_rowmin_kernel<false><<<(strips(N) + 3) / 4, BD, 0, stream>>>(sv, N, gp, G, m_s_gt, nullptr);
    cdist_rowmin_kernel<false><<<(strips(G) + 3) / 4, BD, 0, stream>>>(gp, G, sv, N, m_gt_s, nullptr);
    // nearest teacher vertex (argmin)
    cdist_rowmin_kernel<true><<<(strips(N) + 3) / 4, BD, 0, stream>>>(sv, N, tv, M, m_s_tv, nearest);

    // chamfer means (sum of sqrt of clamped min-d2)
    sum_sqrt_kernel<<<(N + 255) / 256, 256, 0, stream>>>(m_s_tp, N, accum + 0);
    sum_sqrt_kernel<<<(P + 255) / 256, 256, 0, stream>>>(m_tp_s, P, accum + 1);
    sum_sqrt_kernel<<<(N + 255) / 256, 256, 0, stream>>>(m_s_gt, N, accum + 2);
    sum_sqrt_kernel<<<(G + 255) / 256, 256, 0, stream>>>(m_gt_s, G, accum + 3);

    // edge loss + adjacency
    edge_kernel<<<(F + 255) / 256, 256, 0, stream>>>(sv, faces, F, nbr, deg, accum + 4);
    // laplacian smoothing
    smooth_kernel<<<(N + 255) / 256, 256, 0, stream>>>(sv, nbr, deg, N, accum + 5);
    // normal cosine
    normal_kernel<<<(N + 255) / 256, 256, 0, stream>>>(snrm, tnrm, nearest, N, accum + 6);

    finalize_kernel<<<1, 1, 0, stream>>>(accum, N, P, G, F, (float*)d_out);
}